// Consistent_loss_up_4_25288767439317
// MI455X (gfx1250) — compile-verified
//
#include <hip/hip_runtime.h>
#include <stdint.h>

#define GRID_N   256
#define CELLS    (GRID_N * GRID_N)   // 65536
#define HALF_N   (CELLS / 2)         // 32768
#define NBATCH   256
#define BLOCK    1024                // 32 wave32 waves
#define CHUNK    4096                // floats per staged chunk (16 KB)
#define NCHUNK   (CELLS / CHUNK)     // 16

// ---- gfx1250 async global->LDS copy (ASYNCcnt path) ----
// Per-lane 16B copy: 16 bytes from gaddr into LDS at lds_byte_off. Tracked by ASYNCcnt.
// s_wait_dscnt 0 first: async LDS writes are unordered vs DS reads of the
// previously-consumed buffer (WAR on double-buffer reuse).
__device__ __forceinline__ void async_ld16(uint32_t lds_byte_off, const void* gaddr) {
    asm volatile("s_wait_dscnt 0x0\n\t"
                 "global_load_async_to_lds_b128 %0, %1, off"
                 :: "v"(lds_byte_off), "v"(gaddr)
                 : "memory");
}
#define WAIT_ASYNC(N) asm volatile("s_wait_asynccnt " #N ::: "memory")

__device__ __forceinline__ uint32_t lds_offset_of(const void* p) {
    // Flat pointers into the LDS aperture carry the wave-relative LDS byte
    // address in their low 32 bits (CDNA5 ISA 10.2 aperture mapping).
    return (uint32_t)(uintptr_t)p;
}

// One workgroup per (batch, side). side 0: rows 0..127 vs left; side 1: rows 128..255 vs right.
__global__ void __launch_bounds__(BLOCK)
scatter_compare_kernel(const float* __restrict__ up,
                       const float* __restrict__ left,
                       const float* __restrict__ right,
                       float* __restrict__ partials)
{
    extern __shared__ __align__(16) char smem[];
    float* sgrid = (float*)smem;                              // 256 KB scatter grid
    float* stage = (float*)(smem + (size_t)CELLS * 4);        // 2 x 16 KB staging
    float* sred  = (float*)(smem + (size_t)CELLS * 4 + 2 * CHUNK * 4); // 32 floats

    const int wg   = blockIdx.x;
    const int b    = wg >> 1;
    const int side = wg & 1;
    const int tid  = threadIdx.x;

    // ---- phase 0: zero the 256 KB grid (ds_store_b128) ----
    float4* g4 = (float4*)sgrid;
    const float4 z4 = make_float4(0.f, 0.f, 0.f, 0.f);
#pragma unroll
    for (int k = 0; k < CELLS / 4 / BLOCK; ++k)   // 16 stores/thread
        g4[tid + k * BLOCK] = z4;
    __syncthreads();

    // ---- phase 1: scatter-max into LDS via native ds_max (values >= 0 ->
    //      integer max == float max on bit patterns) ----
    const float4* up4 = (const float4*)(up + (size_t)b * CELLS + (size_t)side * HALF_N);
#pragma unroll
    for (int k = 0; k < HALF_N / 4 / BLOCK; ++k) { // 8 float4/thread
        const int v = tid + k * BLOCK;
        const float4 u = up4[v];
        const float uv[4] = {u.x, u.y, u.z, u.w};
        const int p0 = v * 4;
#pragma unroll
        for (int e = 0; e < 4; ++e) {
            const int p = p0 + e;
            const int r = p >> 8;      // local row 0..127
            const int j = p & 255;     // column
            float fz = rintf(uv[e] * 50.f + 110.f);          // round-half-even, like jnp.round
            fz = fminf(fmaxf(fz, 0.f), 255.f);
            const int zi = (int)fz;
            // side 0: (128 - i)/60 with i = r; side 1: (i - 128)/60 with i = r + 128
            const float val = side ? (float)r * (1.f / 60.f)
                                   : (float)(128 - r) * (1.f / 60.f);
            atomicMax((int*)&sgrid[j * GRID_N + zi], __float_as_int(val));
        }
    }
    __syncthreads();

    // ---- phase 2: double-buffered async global->LDS stage of compare data,
    //      then masked-|diff| accumulate against the LDS grid ----
    const float* cmp = (side ? right : left) + (size_t)b * CELLS;
    async_ld16(lds_offset_of(&stage[tid * 4]), cmp + (size_t)tid * 4);  // chunk 0

    float acc = 0.f;
    for (int c = 0; c < NCHUNK; ++c) {
        const int buf = c & 1;
        if (c + 1 < NCHUNK) {
            async_ld16(lds_offset_of(&stage[(1 - buf) * CHUNK + tid * 4]),
                       cmp + (size_t)(c + 1) * CHUNK + (size_t)tid * 4);
            WAIT_ASYNC(0x1);   // chunk c complete (in-order), chunk c+1 in flight
        } else {
            WAIT_ASYNC(0x0);
        }
        const float4 cv = *(const float4*)&stage[buf * CHUNK + tid * 4];
        const float4 uv = *(const float4*)&sgrid[c * CHUNK + tid * 4];
        const float cc[4] = {cv.x, cv.y, cv.z, cv.w};
        const float uu[4] = {uv.x, uv.y, uv.z, uv.w};
#pragma unroll
        for (int e = 0; e < 4; ++e) {
            const float d = fabsf(uu[e] - cc[e]);
            acc += ((d < 0.2f) && (uu[e] != 0.f)) ? d : 0.f;
        }
    }

    // ---- workgroup reduction (wave32 shuffle tree, then cross-wave in LDS) ----
#pragma unroll
    for (int off = 16; off > 0; off >>= 1)
        acc += __shfl_down(acc, off, 32);
    const int lane = tid & 31;
    const int wid  = tid >> 5;
    if (lane == 0) sred[wid] = acc;
    __syncthreads();
    if (wid == 0) {
        float s = (lane < BLOCK / 32) ? sred[lane] : 0.f;
#pragma unroll
        for (int off = 16; off > 0; off >>= 1)
            s += __shfl_down(s, off, 32);
        if (lane == 0) partials[wg] = s;
    }
}

// Deterministic final reduction of the 512 per-workgroup partials.
__global__ void __launch_bounds__(512)
final_reduce_kernel(const float* __restrict__ partials, float* __restrict__ out)
{
    __shared__ float s[512];
    const int t = threadIdx.x;
    s[t] = partials[t];
    __syncthreads();
#pragma unroll
    for (int off = 256; off > 0; off >>= 1) {
        if (t < off) s[t] += s[t + off];
        __syncthreads();
    }
    if (t == 0)
        out[0] = s[0] * (1.f / ((float)NBATCH * (float)CELLS));
}

extern "C" void kernel_launch(void* const* d_in, const int* in_sizes, int n_in,
                              void* d_out, int out_size, void* d_ws, size_t ws_size,
                              hipStream_t stream)
{
    const float* up    = (const float*)d_in[0];
    const float* left  = (const float*)d_in[1];
    const float* right = (const float*)d_in[2];
    float* out      = (float*)d_out;
    float* partials = (float*)d_ws;   // 512 floats

    const size_t shmem = (size_t)CELLS * 4      // 256 KB grid
                       + (size_t)2 * CHUNK * 4  // 32 KB staging
                       + 32 * 4;                // reduction scratch
    // Opt in to >64KB dynamic LDS (idempotent; safe under graph capture).
    (void)hipFuncSetAttribute((const void*)scatter_compare_kernel,
                              hipFuncAttributeMaxDynamicSharedMemorySize,
                              (int)shmem);

    scatter_compare_kernel<<<NBATCH * 2, BLOCK, shmem, stream>>>(up, left, right, partials);
    final_reduce_kernel<<<1, 512, 0, stream>>>(partials, out);
}